// ClusterGCNLayer_14705968021777
// MI455X (gfx1250) — compile-verified
//
#include <hip/hip_runtime.h>

typedef __attribute__((ext_vector_type(2))) float v2f;
typedef __attribute__((ext_vector_type(8))) float v8f;

// ---------------------------------------------------------------------------
// Kernel 1: zero a contiguous float region (agg | deg | ccnt), grid-stride.
// ---------------------------------------------------------------------------
__global__ __launch_bounds__(256) void zero_kernel(float* __restrict__ p, size_t n) {
    size_t i = (size_t)blockIdx.x * blockDim.x + threadIdx.x;
    size_t stride = (size_t)gridDim.x * blockDim.x;
    for (; i < n; i += stride) p[i] = 0.0f;
}

// ---------------------------------------------------------------------------
// Kernel 2: per-edge intra mask -> degree (global f32 atomics) and per-cluster
// edge counts (LDS-aggregated first, then one flush per block).
// ---------------------------------------------------------------------------
__global__ __launch_bounds__(256) void edge_deg_kernel(
    const int* __restrict__ ei, const int* __restrict__ ca,
    float* __restrict__ deg, float* __restrict__ ccnt, int E) {
    __shared__ float lcnt[256];
    lcnt[threadIdx.x] = 0.0f;
    __syncthreads();
    int e = blockIdx.x * 256 + threadIdx.x;
    if (e < E) {
        int s = ei[e];
        int d = ei[E + e];
        int cs = ca[s];
        int cd = ca[d];
        if (cs == cd) {
            unsafeAtomicAdd(&deg[d], 1.0f);          // global_atomic_add_f32
            atomicAdd(&lcnt[cd & 255], 1.0f);        // ds_add_f32
        }
    }
    __syncthreads();
    float c = lcnt[threadIdx.x];
    if (c != 0.0f) unsafeAtomicAdd(&ccnt[threadIdx.x], c);
}

// ---------------------------------------------------------------------------
// Kernel 3: deg -> dinv = rsqrt(deg + 1) in place.
// ---------------------------------------------------------------------------
__global__ __launch_bounds__(256) void node_prep_kernel(float* __restrict__ deg, int N) {
    int n = blockIdx.x * 256 + threadIdx.x;
    if (n < N) deg[n] = rsqrtf(deg[n] + 1.0f);
}

// ---------------------------------------------------------------------------
// Kernel 4: Xl = X @ W via V_WMMA_F32_16X16X4_F32 (exact fp32, memory-bound
// GEMM so no reason to drop precision). D is fixed at 256.
// Block = 256 threads = 8 waves; block owns 16 output rows; wave w owns
// columns [32w, 32w+32) (two 16-col tiles). K loop: 64 steps of k+=4.
// ---------------------------------------------------------------------------
__global__ __launch_bounds__(256) void gemm_wmma_f32(
    const float* __restrict__ X, const float* __restrict__ W,
    float* __restrict__ Xl, int N) {
    const int wave = threadIdx.x >> 5;
    const int lane = threadIdx.x & 31;
    const int half = lane >> 4;      // 0: lanes 0-15, 1: lanes 16-31
    const int l16  = lane & 15;
    const int row0 = blockIdx.x << 4;
    const int c0   = wave << 5;      // two 16-col tiles: c0 and c0+16

    v8f acc0 = {0.f, 0.f, 0.f, 0.f, 0.f, 0.f, 0.f, 0.f};
    v8f acc1 = {0.f, 0.f, 0.f, 0.f, 0.f, 0.f, 0.f, 0.f};

    // A layout: lanes 0-15 hold K=k,k+1 ; lanes 16-31 hold K=k+2,k+3 (per lane float2)
    const float* Xr = X + (size_t)(row0 + l16) * 256 + 2 * half;
    // B layout: v0 = row (k + 2*half), v1 = row (k + 2*half + 1), N = l16
    const float* Wb = W + (size_t)(2 * half) * 256 + l16;

#pragma unroll 8
    for (int k = 0; k < 256; k += 4) {
        float2 at = *(const float2*)(Xr + k);
        v2f a; a.x = at.x; a.y = at.y;
        const float* wk = Wb + (size_t)k * 256;
        v2f b0, b1;
        b0.x = wk[c0];            b0.y = wk[256 + c0];
        b1.x = wk[c0 + 16];       b1.y = wk[256 + c0 + 16];
        acc0 = __builtin_amdgcn_wmma_f32_16x16x4_f32(
            false, a, false, b0, (short)0, acc0, false, false);
        acc1 = __builtin_amdgcn_wmma_f32_16x16x4_f32(
            false, a, false, b1, (short)0, acc1, false, false);
    }

    // D layout: VGPR i -> row i (lanes 0-15) / row 8+i (lanes 16-31), col = l16
    float* o = Xl + (size_t)row0 * 256 + c0 + l16;
#pragma unroll
    for (int i = 0; i < 8; ++i) {
        int r = i + (half << 3);
        o[(size_t)r * 256]      = acc0[i];
        o[(size_t)r * 256 + 16] = acc1[i];
    }
}

// ---------------------------------------------------------------------------
// Kernel 5: edge scatter. agg[dst] += dinv[src]*dinv[dst] * Xl[src].
// 64 lanes per edge, float4 gather, 4 hw f32 atomics per lane (L2-resident).
// ---------------------------------------------------------------------------
__global__ __launch_bounds__(256) void scatter_kernel(
    const int* __restrict__ ei, const int* __restrict__ ca,
    const float* __restrict__ dinv, const float* __restrict__ Xl,
    float* __restrict__ agg, int E) {
    int e = (blockIdx.x << 2) + (threadIdx.x >> 6);
    if (e >= E) return;
    int t = threadIdx.x & 63;
    int s = ei[e];
    int d = ei[E + e];
    if (ca[s] != ca[d]) return;
    float nrm = dinv[s] * dinv[d];
    const float4 v = *(const float4*)(Xl + ((size_t)s << 8) + (t << 2));
    float* o = agg + ((size_t)d << 8) + (t << 2);
    unsafeAtomicAdd(o + 0, nrm * v.x);
    unsafeAtomicAdd(o + 1, nrm * v.y);
    unsafeAtomicAdd(o + 2, nrm * v.z);
    unsafeAtomicAdd(o + 3, nrm * v.w);
}

// ---------------------------------------------------------------------------
// Kernel 6: out = active ? agg + dinv^2 * Xl + bias : X   (float4 per thread)
// ---------------------------------------------------------------------------
__global__ __launch_bounds__(256) void finalize_kernel(
    const float4* __restrict__ X4, const float4* __restrict__ Xl4,
    const float4* __restrict__ agg4, const float* __restrict__ dinv,
    const float* __restrict__ ccnt, const int* __restrict__ ca,
    const float4* __restrict__ bias4, float4* __restrict__ out4,
    long long total) {
    long long i = (long long)blockIdx.x * 256 + threadIdx.x;
    if (i >= total) return;
    int n = (int)(i >> 6);   // 64 float4 chunks per node (D=256)
    int c = (int)(i & 63);
    if (ccnt[ca[n]] > 0.0f) {
        float di = dinv[n];
        float sc = di * di;
        float4 xl = Xl4[i];
        float4 ag = agg4[i];
        float4 b  = bias4[c];
        float4 r;
        r.x = ag.x + sc * xl.x + b.x;
        r.y = ag.y + sc * xl.y + b.y;
        r.z = ag.z + sc * xl.z + b.z;
        r.w = ag.w + sc * xl.w + b.w;
        out4[i] = r;
    } else {
        out4[i] = X4[i];
    }
}

// ---------------------------------------------------------------------------
// Host-side launch sequence (single stream => implicit ordering).
// inputs: X[N*256] f32, weight[256*256] f32, bias[256] f32,
//         cluster_assignment[N] i32, edge_index[2*E] i32
// ---------------------------------------------------------------------------
extern "C" void kernel_launch(void* const* d_in, const int* in_sizes, int n_in,
                              void* d_out, int out_size, void* d_ws, size_t ws_size,
                              hipStream_t stream) {
    const float* X    = (const float*)d_in[0];
    const float* W    = (const float*)d_in[1];
    const float* bias = (const float*)d_in[2];
    const int*   ca   = (const int*)d_in[3];
    const int*   ei   = (const int*)d_in[4];
    const int N = in_sizes[3];
    const int E = in_sizes[4] / 2;
    float* out = (float*)d_out;

    // Workspace carve-out: [ Xl | agg | deg/dinv | ccnt(256) ]
    float* Xl   = (float*)d_ws;                 // N*256
    float* agg  = Xl  + (size_t)N * 256;        // N*256
    float* deg  = agg + (size_t)N * 256;        // N (becomes dinv)
    float* ccnt = deg + N;                      // 256

    // 1) zero agg, deg, ccnt (they are contiguous)
    size_t zcount = (size_t)N * 256 + (size_t)N + 256;
    zero_kernel<<<2048, 256, 0, stream>>>(agg, zcount);

    // 2) degrees + per-cluster intra-edge counts
    edge_deg_kernel<<<(E + 255) / 256, 256, 0, stream>>>(ei, ca, deg, ccnt, E);

    // 3) dinv = rsqrt(deg + 1)
    node_prep_kernel<<<(N + 255) / 256, 256, 0, stream>>>(deg, N);

    // 4) Xl = X @ W  (WMMA f32; N is a multiple of 16)
    gemm_wmma_f32<<<N / 16, 256, 0, stream>>>(X, W, Xl, N);

    // 5) edge scatter into agg
    scatter_kernel<<<(E + 3) / 4, 256, 0, stream>>>(ei, ca, deg, Xl, agg, E);

    // 6) combine + cluster-active select
    long long total = (long long)N * 64;  // float4 elements
    finalize_kernel<<<(unsigned)((total + 255) / 256), 256, 0, stream>>>(
        (const float4*)X, (const float4*)Xl, (const float4*)agg,
        deg, ccnt, ca, (const float4*)bias, (float4*)out, total);
}